// GraphConv_18339510354235
// MI455X (gfx1250) — compile-verified
//
#include <hip/hip_runtime.h>
#include <hip/hip_bf16.h>

// ---------------------------------------------------------------------------
// GraphConv (GCN) for MI455X / gfx1250, wave32.
//   deg_out = clamp(segsum(1, src), 1);  feat_s = feat * rsqrt(deg_out)
//   agg[dst] += feat_s[src]              (scatter-add, L2-resident)
//   out = (agg @ W^T + b) * rsqrt(clamp(deg_in,1))   (fp32 WMMA 16x16x4)
// ---------------------------------------------------------------------------

#define N_NODES 50000
#define N_EDGES 800000
#define IN_DIM  128
#define OUT_DIM 128

typedef __attribute__((ext_vector_type(2))) float v2f;
typedef __attribute__((ext_vector_type(8))) float v8f;

// ---------------------------------------------------------------- zero fill
__global__ __launch_bounds__(256) void gc_zero_kernel(float* __restrict__ p, int n) {
    int i = blockIdx.x * 256 + threadIdx.x;
    if (i < n) p[i] = 0.0f;
}

// ------------------------------------------------------------- degree count
__global__ __launch_bounds__(256) void gc_degree_kernel(const int* __restrict__ src,
                                                        const int* __restrict__ dst,
                                                        float* __restrict__ deg_out,
                                                        float* __restrict__ deg_in) {
    int e = blockIdx.x * 256 + threadIdx.x;
    if (e >= N_EDGES) return;
    atomicAdd(&deg_out[src[e]], 1.0f);
    atomicAdd(&deg_in[dst[e]], 1.0f);
}

// ----------------------------------------------- edge scatter (1 wave/edge)
// lane l handles floats [4l .. 4l+3] of the 128-float row.
__global__ __launch_bounds__(256) void gc_scatter_kernel(const float* __restrict__ feat,
                                                         const int* __restrict__ src,
                                                         const int* __restrict__ dst,
                                                         const float* __restrict__ deg_out,
                                                         float* __restrict__ agg) {
    int lane = threadIdx.x & 31;
    int wave = threadIdx.x >> 5;
    int e = blockIdx.x * 8 + wave;
    if (e >= N_EDGES) return;

    int s = src[e];
    int d = dst[e];
    float dv = deg_out[s];
    float scale = rsqrtf(dv < 1.0f ? 1.0f : dv);

    const float4 f = *((const float4*)(feat + (size_t)s * IN_DIM) + lane);
    float* ap = agg + (size_t)d * IN_DIM + lane * 4;
    atomicAdd(ap + 0, f.x * scale);
    atomicAdd(ap + 1, f.y * scale);
    atomicAdd(ap + 2, f.z * scale);
    atomicAdd(ap + 3, f.w * scale);
}

// --------------------------------------------- WMMA fp32 GEMM (16x16 tiles)
// One wave computes a 16x16 tile of out = agg @ W^T, K-loop in steps of 4
// using V_WMMA_F32_16X16X4_F32.
//
// A (16x4, f32):  lane L (L<16): row M=L, holds K={k,k+1}; lanes 16-31 hold
//                 K={k+2,k+3} for the same rows.  (ISA 7.12.2)
// B (4x16, f32):  lane L: column N=L%16, VGPR0/1 hold the same K pair as A.
// C/D (16x16):    VGPR v, lane L -> element [M = v + (L>>4)*8][N = L%16].
__global__ __launch_bounds__(256) void gc_gemm_kernel(const float* __restrict__ agg,
                                                      const float* __restrict__ W,
                                                      const float* __restrict__ bias,
                                                      const float* __restrict__ deg_in,
                                                      float* __restrict__ out) {
    const int lane = threadIdx.x & 31;
    const int wave = threadIdx.x >> 5;
    const int tile = blockIdx.x * 8 + wave;          // 3125*8 = 25000 tiles
    const int mTile = tile >> 3;                     // 3125 row tiles
    const int nTile = tile & 7;                      // 8 col tiles
    const int m0 = mTile * 16;
    const int n0 = nTile * 16;
    if (m0 >= N_NODES) return;

    const int half = lane >> 4;                      // 0: K pair {0,1}; 1: {2,3}
    const int lid  = lane & 15;
    const int arow = m0 + lid;                       // A-matrix row (node)
    const int bcol = n0 + lid;                       // B-matrix col (out channel)

    const float* aptr = agg + (size_t)arow * IN_DIM + half * 2;
    const float* bptr = W   + (size_t)bcol * IN_DIM + half * 2;

    v8f c = {0.f, 0.f, 0.f, 0.f, 0.f, 0.f, 0.f, 0.f};
    #pragma unroll
    for (int k = 0; k < IN_DIM; k += 4) {
        v2f a = *(const v2f*)(aptr + k);
        v2f b = *(const v2f*)(bptr + k);
        // 8 args: (neg_a, A, neg_b, B, c_mod, C, reuse_a, reuse_b)
        c = __builtin_amdgcn_wmma_f32_16x16x4_f32(false, a, false, b,
                                                  (short)0, c, false, false);
    }

    const float bn = bias[bcol];
    #pragma unroll
    for (int v = 0; v < 8; ++v) {
        const int m = m0 + v + half * 8;
        float dv = deg_in[m];
        float s = rsqrtf(dv < 1.0f ? 1.0f : dv);
        out[(size_t)m * OUT_DIM + bcol] = (c[v] + bn) * s;
    }
}

// ---------------------------------------------------------------------------
extern "C" void kernel_launch(void* const* d_in, const int* in_sizes, int n_in,
                              void* d_out, int out_size, void* d_ws, size_t ws_size,
                              hipStream_t stream) {
    const float* feat = (const float*)d_in[0];
    const int*   src  = (const int*)d_in[1];
    const int*   dst  = (const int*)d_in[2];
    const float* W    = (const float*)d_in[3];
    const float* bias = (const float*)d_in[4];
    float*       out  = (float*)d_out;

    // Workspace layout: agg [N*IN], deg_out [N], deg_in [N]
    float* agg     = (float*)d_ws;
    float* deg_out = agg + (size_t)N_NODES * IN_DIM;
    float* deg_in  = deg_out + N_NODES;

    const int n_zero = N_NODES * IN_DIM + 2 * N_NODES;   // 6.5M floats
    gc_zero_kernel<<<(n_zero + 255) / 256, 256, 0, stream>>>(agg, n_zero);

    gc_degree_kernel<<<(N_EDGES + 255) / 256, 256, 0, stream>>>(src, dst, deg_out, deg_in);

    // 8 edges per 256-thread block (1 wave per edge)
    gc_scatter_kernel<<<N_EDGES / 8, 256, 0, stream>>>(feat, src, dst, deg_out, agg);

    // 25000 tiles / 8 waves per block
    gc_gemm_kernel<<<(N_NODES / 16) * (OUT_DIM / 16) / 8, 256, 0, stream>>>(
        agg, W, bias, deg_in, out);
}